// QNetBlock_40106404610627
// MI455X (gfx1250) — compile-verified
//
#include <hip/hip_runtime.h>
#include <math.h>

#define NW      18
#define NSTATE  (1 << NW)          // 262144 amplitudes per batch element
#define BATCH   32
#define NQL     2

typedef __attribute__((ext_vector_type(2))) float v2f;
typedef __attribute__((ext_vector_type(8))) float v8f;

struct PermMasks { unsigned m[NW]; };   // gather map: srcBit[p] = parity(idx & m[p])
struct GateTbl   { unsigned f[14]; unsigned w[4]; }; // permuted images of index bits
struct BitImg    { unsigned img[NW]; };              // permuted image of each bit

// -------------------------------------------------------------------------
// Per-batch L2 norm (one block per batch element)
// -------------------------------------------------------------------------
__global__ void k_norm(const float* __restrict__ x, float* __restrict__ norms) {
  int b = blockIdx.x;
  const float* p = x + (size_t)b * NSTATE;
  float s = 0.f;
  for (int i = threadIdx.x; i < NSTATE; i += blockDim.x) { float v = p[i]; s += v * v; }
  __shared__ float red[256];
  red[threadIdx.x] = s;
  __syncthreads();
  for (int off = 128; off > 0; off >>= 1) {
    if (threadIdx.x < off) red[threadIdx.x] += red[threadIdx.x + off];
    __syncthreads();
  }
  if (threadIdx.x == 0) norms[b] = rsqrtf(red[0]);
}

// -------------------------------------------------------------------------
// stateR = x * rsqrt(sumsq), stateI = 0   (planar complex, SoA)
// -------------------------------------------------------------------------
__global__ void k_scale(const float* __restrict__ x, const float* __restrict__ norms,
                        float* __restrict__ sR, float* __restrict__ sI) {
  size_t j = (size_t)blockIdx.x * blockDim.x + threadIdx.x;
  int b = (int)(j >> NW);
  float n = norms[b];
  sR[j] = x[j] * n;
  sI[j] = 0.f;
}

// -------------------------------------------------------------------------
// Build fused 16x16 complex group unitaries: Kronecker of four 2x2 Rot gates
// (group 4 = U16 (x) U17 (x) I (x) I).  One block, 256 threads.
// -------------------------------------------------------------------------
__global__ void k_build(const float* __restrict__ params,
                        float* __restrict__ Mr, float* __restrict__ Mi) {
  __shared__ float Ur[NQL][NW][2][2], Ui[NQL][NW][2][2];
  int t = threadIdx.x;
  if (t < NQL * NW) {
    int l = t / NW, w = t % NW;
    float phi = params[(l * NW + w) * 3 + 0];
    float th  = params[(l * NW + w) * 3 + 1];
    float om  = params[(l * NW + w) * 3 + 2];
    float c = cosf(th * 0.5f), s = sinf(th * 0.5f);
    float a0 = -(phi + om) * 0.5f;   // m00 =  e^{i a0} c
    float a1 =  (phi - om) * 0.5f;   // m01 = -e^{i a1} s
    float a2 = -(phi - om) * 0.5f;   // m10 =  e^{i a2} s
    float a3 =  (phi + om) * 0.5f;   // m11 =  e^{i a3} c
    Ur[l][w][0][0] =  cosf(a0) * c;  Ui[l][w][0][0] =  sinf(a0) * c;
    Ur[l][w][0][1] = -cosf(a1) * s;  Ui[l][w][0][1] = -sinf(a1) * s;
    Ur[l][w][1][0] =  cosf(a2) * s;  Ui[l][w][1][0] =  sinf(a2) * s;
    Ur[l][w][1][1] =  cosf(a3) * c;  Ui[l][w][1][1] =  sinf(a3) * c;
  }
  __syncthreads();
  int i = t >> 4, j = t & 15;        // one (i,j) entry per thread
  for (int l = 0; l < NQL; ++l) {
    for (int g = 0; g < 5; ++g) {
      int w4[4];
      if (g < 4) { w4[0] = 4*g; w4[1] = 4*g+1; w4[2] = 4*g+2; w4[3] = 4*g+3; }
      else       { w4[0] = 16;  w4[1] = 17;    w4[2] = -1;    w4[3] = -1;    }
      float re = 1.f, im = 0.f;
#pragma unroll
      for (int k = 0; k < 4; ++k) {
        int ib = (i >> (3 - k)) & 1, jb = (j >> (3 - k)) & 1;
        float fr, fi;
        if (w4[k] < 0) { fr = (ib == jb) ? 1.f : 0.f; fi = 0.f; }
        else           { fr = Ur[l][w4[k]][ib][jb]; fi = Ui[l][w4[k]][ib][jb]; }
        float nr = re * fr - im * fi;
        float ni = re * fi + im * fr;
        re = nr; im = ni;
      }
      Mr[(l * 5 + g) * 256 + t] = re;
      Mi[(l * 5 + g) * 256 + t] = im;
    }
  }
}

// -------------------------------------------------------------------------
// Fused 4-wire gate pass: one wave per 16x16 complex tile, in place.
//   D = M * S  via chained V_WMMA_F32_16X16X4_F32 (K accumulated 4 at a time)
//   Dr = Mr*Sr + (-Mi)*Si ; Di = Mr*Si + Mi*Sr   (16 WMMAs / tile)
// All GF(2) permutation algebra is pre-solved on the host into bit-images:
//   addr = XOR over set bits of (outer index, row index) of their images.
// -------------------------------------------------------------------------
__global__ void k_gate(float* __restrict__ sR, float* __restrict__ sI,
                       const float* __restrict__ Mr, const float* __restrict__ Mi,
                       GateTbl tb) {
  unsigned lane = threadIdx.x & 31;
  unsigned tile = blockIdx.x * (blockDim.x >> 5) + (threadIdx.x >> 5); // 0..32767
  unsigned b    = tile >> 10;                       // 1024 outer-tiles / batch
  unsigned ot   = tile & 1023;
  unsigned half = lane >> 4;
  unsigned col  = lane & 15;

  // physical base address for this lane's column (14-bit outer index)
  unsigned o = ot * 16u + col;
  unsigned pbase = 0;
#pragma unroll
  for (int k = 0; k < 14; ++k)
    pbase ^= ((o >> k) & 1u) ? tb.f[k] : 0u;

  auto paddr = [&](unsigned r) -> unsigned {
    return pbase ^ (((r >> 3) & 1u) ? tb.w[0] : 0u) ^ (((r >> 2) & 1u) ? tb.w[1] : 0u)
                 ^ (((r >> 1) & 1u) ? tb.w[2] : 0u) ^ ((r & 1u) ? tb.w[3] : 0u);
  };

  size_t boff = (size_t)b << NW;
  float* pR = sR + boff;
  float* pI = sI + boff;

  // Load operands.
  //   B chunk k (4x16): lanes 0-15 hold S[4k+0..1][col], lanes 16-31 S[4k+2..3][col]
  //   A chunk k (16x4): lanes 0-15 hold M[lane][4k+0..1],  lanes 16-31 M[lane-16][4k+2..3]
  v2f br[4], bi[4], ar[4], ai[4], an[4];
#pragma unroll
  for (int k = 0; k < 4; ++k) {
    unsigned r0 = 4u * k + 2u * half, r1 = r0 + 1u;
    unsigned a0 = paddr(r0), a1 = paddr(r1);
    v2f t;
    t[0] = pR[a0]; t[1] = pR[a1]; br[k] = t;
    t[0] = pI[a0]; t[1] = pI[a1]; bi[k] = t;
    unsigned c0 = 4u * k + 2u * half;
    t[0] = Mr[col * 16 + c0];  t[1] = Mr[col * 16 + c0 + 1];  ar[k] = t;
    t[0] = Mi[col * 16 + c0];  t[1] = Mi[col * 16 + c0 + 1];  ai[k] = t;
    t[0] = -ai[k][0];          t[1] = -ai[k][1];              an[k] = t;
  }

  v8f dR = {}, dI = {};
#pragma unroll
  for (int k = 0; k < 4; ++k)
    dR = __builtin_amdgcn_wmma_f32_16x16x4_f32(false, ar[k], false, br[k], (short)0, dR, false, false);
#pragma unroll
  for (int k = 0; k < 4; ++k)
    dR = __builtin_amdgcn_wmma_f32_16x16x4_f32(false, an[k], false, bi[k], (short)0, dR, false, false);
#pragma unroll
  for (int k = 0; k < 4; ++k)
    dI = __builtin_amdgcn_wmma_f32_16x16x4_f32(false, ar[k], false, bi[k], (short)0, dI, false, false);
#pragma unroll
  for (int k = 0; k < 4; ++k)
    dI = __builtin_amdgcn_wmma_f32_16x16x4_f32(false, ai[k], false, br[k], (short)0, dI, false, false);

  // Store: D VGPR r -> row (r + 8*half), column col
#pragma unroll
  for (int r = 0; r < 8; ++r) {
    unsigned a = paddr((unsigned)r + 8u * half);
    pR[a] = dR[r];
    pI[a] = dI[r];
  }
}

// -------------------------------------------------------------------------
// out = |state| gathered through the total accumulated permutation
// -------------------------------------------------------------------------
__global__ void k_abs(const float* __restrict__ sR, const float* __restrict__ sI,
                      float* __restrict__ out, BitImg bi) {
  size_t j = (size_t)blockIdx.x * blockDim.x + threadIdx.x;
  unsigned idx = (unsigned)(j & (NSTATE - 1));
  size_t boff = j & ~(size_t)(NSTATE - 1);
  unsigned src = 0;
#pragma unroll
  for (int p = 0; p < NW; ++p)
    src ^= ((idx >> p) & 1u) ? bi.img[p] : 0u;
  float r = sR[boff + src], i = sI[boff + src];
  out[j] = sqrtf(r * r + i * i);
}

// -------------------------------------------------------------------------
// Host-side GF(2) algebra for the CNOT chains
// -------------------------------------------------------------------------
static PermMasks identity_masks() {
  PermMasks r;
  for (int p = 0; p < NW; ++p) r.m[p] = 1u << p;
  return r;
}
// Gather map of one layer's CNOT ring: CNOT((i+l)%18, (i+l+1)%18), i=0..17
static PermMasks layer_masks(int l) {
  PermMasks r = identity_masks();
  for (int i = 0; i < NW; ++i) {
    int c = (i + l) % NW, t = (i + l + 1) % NW;
    int pc = 17 - c, pt = 17 - t;        // wire w lives at bit position 17-w
    for (int p = 0; p < NW; ++p)
      if ((r.m[p] >> pt) & 1) r.m[p] ^= 1u << pc;
  }
  return r;
}
// (A o B)(j) = A(B(j))
static PermMasks compose(const PermMasks& A, const PermMasks& B) {
  PermMasks r;
  for (int p = 0; p < NW; ++p) {
    unsigned acc = 0;
    for (int q = 0; q < NW; ++q)
      if ((A.m[p] >> q) & 1) acc ^= B.m[q];
    r.m[p] = acc;
  }
  return r;
}
// image of a single bit e_p under the gather map
static unsigned perm_image(const PermMasks& P, unsigned e) {
  unsigned r = 0;
  for (int q = 0; q < NW; ++q)
    r |= (unsigned)(__builtin_popcount(e & P.m[q]) & 1) << q;
  return r;
}

// -------------------------------------------------------------------------
extern "C" void kernel_launch(void* const* d_in, const int* in_sizes, int n_in,
                              void* d_out, int out_size, void* d_ws, size_t ws_size,
                              hipStream_t stream) {
  const float* x      = (const float*)d_in[0];   // (32, 2048, 128) f32
  const float* params = (const float*)d_in[1];   // (2, 18, 3) f32
  float* out = (float*)d_out;

  char* w = (char*)d_ws;
  float* norms = (float*)(w);                    //   128 B
  float* Mr    = (float*)(w + 1024);             // 10*256 f32
  float* Mi    = (float*)(w + 1024 + 10 * 256 * 4);
  size_t SB    = (size_t)BATCH * NSTATE * sizeof(float);   // 32 MB
  float* sR    = (float*)(w + (1 << 15));
  float* sI    = (float*)(w + (1 << 15) + SB);   // total ws use: ~64 MB + 32 KB

  const int NTOT_BLK = (BATCH * NSTATE) / 256;   // 32768

  k_norm <<<BATCH, 256, 0, stream>>>(x, norms);
  k_scale<<<NTOT_BLK, 256, 0, stream>>>(x, norms, sR, sI);
  k_build<<<1, 256, 0, stream>>>(params, Mr, Mi);

  PermMasks G0 = layer_masks(0);
  PermMasks G1 = layer_masks(1);
  PermMasks Gtot = compose(G0, G1);              // abs reads through G0(G1(j))
  PermMasks Id = identity_masks();

  for (int l = 0; l < NQL; ++l) {
    const PermMasks& pm = (l == 0) ? Id : G0;    // layer-1 ops address through G0
    for (int g = 0; g < 5; ++g) {
      int p0, p1, p2, p3;
      if (g < 4) { p0 = 17 - 4*g; p1 = 16 - 4*g; p2 = 15 - 4*g; p3 = 14 - 4*g; }
      else       { p0 = 1; p1 = 0; p2 = 17; p3 = 16; }   // wires {16,17,0,1}, I on 0,1
      unsigned pmask = (1u<<p0)|(1u<<p1)|(1u<<p2)|(1u<<p3);

      GateTbl tb;
      int n = 0;
      for (int bit = 0; bit < NW; ++bit)         // free positions, ascending:
        if (!((pmask >> bit) & 1)) tb.f[n++] = perm_image(pm, 1u << bit);
      tb.w[0] = perm_image(pm, 1u << p0);
      tb.w[1] = perm_image(pm, 1u << p1);
      tb.w[2] = perm_image(pm, 1u << p2);
      tb.w[3] = perm_image(pm, 1u << p3);

      k_gate<<<4096, 256, 0, stream>>>(sR, sI,
                                       Mr + (l * 5 + g) * 256,
                                       Mi + (l * 5 + g) * 256, tb);
    }
  }

  BitImg bi;
  for (int p = 0; p < NW; ++p) bi.img[p] = perm_image(Gtot, 1u << p);
  k_abs<<<NTOT_BLK, 256, 0, stream>>>(sR, sI, out, bi);
}